// buff_Loss_31499290149450
// MI455X (gfx1250) — compile-verified
//
#include <hip/hip_runtime.h>

// MI455X (gfx1250) MSE-style reduction.
// Bandwidth-bound: 256 MiB streamed once -> ~11.5us floor at 23.3 TB/s.
// Strategy: B128 non-temporal loads, grid-stride unrolled x2 so each wave
// keeps 4 B128 loads in flight, prefetch ahead, branchless remap,
// f32 WMMA (16x16x4) all-ones trick for the wave32 reduction, two-pass
// deterministic block-partial reduction through d_ws.

typedef __attribute__((ext_vector_type(2))) float v2f;
typedef __attribute__((ext_vector_type(4))) float v4f;
typedef __attribute__((ext_vector_type(8))) float v8f;

#define SP1 10.0f
#define SP2 5.0f
#define THREADS 256
#define WAVES_PER_BLOCK (THREADS / 32)

// Sum (a0 + a1) over all 32 lanes of the wave, exact f32.
// V_WMMA_F32_16X16X4_F32 with B = all-ones: D[m][n] = sum_k A[m][k]
// (independent of n). Lane L (L<16) holds rows 0..7 of a column in its 8
// D VGPRs, lane L^16 holds rows 8..15, so (local 8-VGPR sum) +
// (same from lane^16) = sum of every A slot = wave-wide sum of (a0+a1).
__device__ __forceinline__ float wave_sum2(float a0, float a1) {
#if __has_builtin(__builtin_amdgcn_wmma_f32_16x16x4_f32)
  v2f a; a[0] = a0;   a[1] = a1;
  v2f b; b[0] = 1.0f; b[1] = 1.0f;   // every element of B == 1 regardless of layout
  v8f c = {};
  v8f d = __builtin_amdgcn_wmma_f32_16x16x4_f32(
      /*neg_a=*/false, a, /*neg_b=*/false, b,
      /*c_mod=*/(short)0, c, /*reuse_a=*/false, /*reuse_b=*/false);
  float s = ((d[0] + d[1]) + (d[2] + d[3])) + ((d[4] + d[5]) + (d[6] + d[7]));
  s += __shfl_xor(s, 16, 32);
  return s;
#else
  float s = a0 + a1;
  #pragma unroll
  for (int off = 16; off > 0; off >>= 1) s += __shfl_xor(s, off, 32);
  return s;
#endif
}

// Remap one element pair and accumulate squared diff.
//   label==1: n*10, l->10 ; label==2: n*5, l->20 ; else unchanged.
__device__ __forceinline__ void accum_one(float n, float l, float& acc) {
  bool m1 = (l == 1.0f);
  bool m2 = (l == 2.0f);
  float sn = m1 ? SP1 : (m2 ? SP2 : 1.0f);
  float lt = (m1 || m2) ? (l * SP1) : l;
  float diff = fmaf(n, sn, -lt);
  acc = fmaf(diff, diff, acc);
}

__device__ __forceinline__ void accum_vec(const v4f& nv, const v4f& lv,
                                          float& a0, float& a1,
                                          float& a2, float& a3) {
  accum_one(nv[0], lv[0], a0);
  accum_one(nv[1], lv[1], a1);
  accum_one(nv[2], lv[2], a2);
  accum_one(nv[3], lv[3], a3);
}

__global__ __launch_bounds__(THREADS) void mse_partial_kernel(
    const float* __restrict__ norms, const float* __restrict__ labels,
    float* __restrict__ partials, long long n4, int tail, float scale) {
  const long long stride  = (long long)gridDim.x * THREADS;
  const long long stride2 = 2 * stride;
  const long long gid = (long long)blockIdx.x * THREADS + threadIdx.x;
  const v4f* __restrict__ np = (const v4f*)norms;
  const v4f* __restrict__ lp = (const v4f*)labels;

  float acc0 = 0.0f, acc1 = 0.0f, acc2 = 0.0f, acc3 = 0.0f;

  // Unrolled x2: 4 independent B128 NT loads in flight per wave per trip.
  long long i = gid;
  for (; i + stride < n4; i += stride2) {
    v4f nv0 = __builtin_nontemporal_load(np + i);            // global_load_b128 NT
    v4f lv0 = __builtin_nontemporal_load(lp + i);
    v4f nv1 = __builtin_nontemporal_load(np + i + stride);
    v4f lv1 = __builtin_nontemporal_load(lp + i + stride);
    // Pull both streams ahead (speculative; OOB translations dropped).
    __builtin_prefetch((const void*)(np + i + 2 * stride2), 0, 0);
    __builtin_prefetch((const void*)(lp + i + 2 * stride2), 0, 0);
    __builtin_prefetch((const void*)(np + i + 2 * stride2 + stride), 0, 0);
    __builtin_prefetch((const void*)(lp + i + 2 * stride2 + stride), 0, 0);
    accum_vec(nv0, lv0, acc0, acc1, acc2, acc3);
    accum_vec(nv1, lv1, acc0, acc1, acc2, acc3);
  }
  if (i < n4) {  // at most one leftover float4 per thread
    v4f nv = __builtin_nontemporal_load(np + i);
    v4f lv = __builtin_nontemporal_load(lp + i);
    accum_vec(nv, lv, acc0, acc1, acc2, acc3);
  }

  if (tail > 0) {  // N % 4 leftovers (none for the 2^25 reference shape)
    long long base = n4 * 4;
    if (gid < tail) accum_one(norms[base + gid], labels[base + gid], acc0);
  }

  // EXEC is all-ones again here (uniform reconvergence) -> WMMA legal.
  float wsum = wave_sum2(acc0 + acc1, acc2 + acc3);

  __shared__ float smem[WAVES_PER_BLOCK];
  const int wave = threadIdx.x >> 5;
  const int lane = threadIdx.x & 31;
  if (lane == 0) smem[wave] = wsum;
  __syncthreads();
  if (threadIdx.x == 0) {
    float b = 0.0f;
    #pragma unroll
    for (int w = 0; w < WAVES_PER_BLOCK; ++w) b += smem[w];
    partials[blockIdx.x] = b * scale;
  }
}

__global__ __launch_bounds__(THREADS) void mse_final_kernel(
    const float* __restrict__ partials, int nparts,
    float* __restrict__ out, float inv_n) {
  float s = 0.0f;
  for (int i = threadIdx.x; i < nparts; i += THREADS) s += partials[i];

  float wsum = wave_sum2(s, 0.0f);

  __shared__ float smem[WAVES_PER_BLOCK];
  const int wave = threadIdx.x >> 5;
  const int lane = threadIdx.x & 31;
  if (lane == 0) smem[wave] = wsum;
  __syncthreads();
  if (threadIdx.x == 0) {
    float t = 0.0f;
    #pragma unroll
    for (int w = 0; w < WAVES_PER_BLOCK; ++w) t += smem[w];
    out[0] = t * inv_n;
  }
}

extern "C" void kernel_launch(void* const* d_in, const int* in_sizes, int n_in,
                              void* d_out, int out_size, void* d_ws, size_t ws_size,
                              hipStream_t stream) {
  (void)n_in; (void)out_size;
  const float* norms  = (const float*)d_in[0];
  const float* labels = (const float*)d_in[1];
  float* out = (float*)d_out;

  const long long N  = (long long)in_sizes[0];
  const long long n4 = N / 4;
  const int tail     = (int)(N % 4);
  const float inv_n  = (float)(1.0 / (double)N);

  int blocks = 2048;                      // 524,288 threads; 16 float4 iters/thread
  const int max_parts = (int)(ws_size / sizeof(float));
  if (blocks > max_parts) blocks = max_parts;

  if (blocks >= 2) {
    float* partials = (float*)d_ws;
    mse_partial_kernel<<<blocks, THREADS, 0, stream>>>(norms, labels, partials,
                                                       n4, tail, 1.0f);
    mse_final_kernel<<<1, THREADS, 0, stream>>>(partials, blocks, out, inv_n);
  } else {
    // Scratch too small for partials: single block writes the mean directly.
    mse_partial_kernel<<<1, THREADS, 0, stream>>>(norms, labels, out,
                                                  n4, tail, inv_n);
  }
}